// SSMBlock_88304527606387
// MI455X (gfx1250) — compile-verified
//
#include <hip/hip_runtime.h>
#include <math.h>

// Problem constants (match reference)
#define B_SZ 4
#define SEQ  4096
#define DM   1024
#define NS   16
#define NROW (B_SZ*SEQ)   // 16384 rows flattened
#define SPLIT 32          // seq splits for mean reduction

typedef __attribute__((ext_vector_type(2))) float v2f;
typedef __attribute__((ext_vector_type(8))) float v8f;

__device__ __forceinline__ v8f wmma_f32_k4(v2f a, v2f b, v8f c) {
  // V_WMMA_F32_16X16X4_F32 : D(16x16,f32) = A(16x4,f32) * B(4x16,f32) + C
  return __builtin_amdgcn_wmma_f32_16x16x4_f32(
      /*neg_a=*/false, a, /*neg_b=*/false, b,
      /*c_mod=*/(short)0, c, /*reuse_a=*/false, /*reuse_b=*/false);
}

// Hardware V_TANH_F32 (CDNA5 trans op) instead of the ~30-op libm expansion.
__device__ __forceinline__ float fast_tanh(float v) {
#if __has_builtin(__builtin_amdgcn_tanhf)
  return __builtin_amdgcn_tanhf(v);
#else
  float r;
  asm volatile("v_tanh_f32 %0, %1\n\tv_nop" : "=v"(r) : "v"(v));
  return r;
#endif
}

// ---------------------------------------------------------------------------
// 1) partial column sums of x over the sequence (deterministic, no atomics)
__global__ __launch_bounds__(256) void mean_kernel(const float* __restrict__ x,
                                                   float* __restrict__ xpart) {
  const int SCHUNK = SEQ / SPLIT;                  // 128 seq rows per block
  int blk = blockIdx.x;                            // b*SPLIT + split
  int b = blk >> 5, split = blk & 31;
  const float* xb = x + ((size_t)b * SEQ + (size_t)split * SCHUNK) * DM;
  int tid = threadIdx.x;
  for (int dc = 0; dc < DM; dc += 256) {
    int d = dc + tid;
    float sum = 0.f;
    for (int s = 0; s < SCHUNK; ++s) sum += xb[(size_t)s * DM + d];
    xpart[(size_t)blk * DM + d] = sum;
  }
}

// 2) deterministic fixed-order reduction of the partials
__global__ __launch_bounds__(256) void reduce_kernel(const float* __restrict__ xpart,
                                                     float* __restrict__ xsum) {
  int i = blockIdx.x * 256 + threadIdx.x;          // 0 .. B_SZ*DM-1
  if (i >= B_SZ * DM) return;
  int b = i / DM, d = i % DM;
  float s = 0.f;
  for (int sp = 0; sp < SPLIT; ++sp)
    s += xpart[((size_t)b * SPLIT + sp) * DM + d];
  xsum[i] = s;
}

// 3) gate MLP: cf[b] = sigmoid( silu(xm@w1^T + b1) @ w2^T + b2 )
__global__ __launch_bounds__(256) void gate_kernel(const float* __restrict__ xsum,
                                                   const float* __restrict__ w1,
                                                   const float* __restrict__ b1,
                                                   const float* __restrict__ w2,
                                                   const float* __restrict__ b2,
                                                   float* __restrict__ cf) {
  __shared__ float red[256];
  int j = threadIdx.x;
  const float inv_s = 1.0f / (float)SEQ;
  for (int b = 0; b < B_SZ; ++b) {
    float g = b1[j];
    const float* xs = xsum + b * DM;
    const float* w  = w1 + (size_t)j * DM;
    for (int d = 0; d < DM; ++d) g = fmaf(xs[d] * inv_s, w[d], g);
    float sg = g / (1.f + __expf(-g));             // SiLU
    red[j] = sg * w2[j];
    __syncthreads();
    for (int off = 128; off > 0; off >>= 1) {
      if (j < off) red[j] += red[j + off];
      __syncthreads();
    }
    if (j == 0) {
      float z = red[0] + b2[0];
      cf[b] = 1.f / (1.f + __expf(-z));
    }
    __syncthreads();
  }
}

// 4) uB[row, n] = tanh( cf[b] * (x[row,:] @ B[n,:]) )  via WMMA f32 16x16x4
//    One wave owns one 16-row tile; B (64KB) transposed into LDS per block.
__global__ __launch_bounds__(128) void ub_gemm_kernel(const float* __restrict__ x,
                                                      const float* __restrict__ Bw,
                                                      const float* __restrict__ cf,
                                                      float* __restrict__ uB) {
  __shared__ float Bt[DM * NS];                    // Bt[k*16+n] = B[n*DM+k]
  int tid = threadIdx.x;
  for (int i = tid; i < DM * NS; i += 128) {
    int n = i / DM, k = i % DM;                    // coalesced read over k
    Bt[k * NS + n] = Bw[i];
  }
  __syncthreads();

  int wave = tid >> 5, lane = tid & 31;
  int tile = blockIdx.x * 4 + wave;                // 0..1023, never crosses batch
  int row0 = tile * 16;
  int b    = row0 / SEQ;
  int mrow = lane & 15;                            // A-frag row / frag col index
  int koff = (lane >> 4) * 2;                      // K sub-offset per half-wave
  const float* xrow = x + (size_t)(row0 + mrow) * DM + koff;

  v8f acc = {};
  #pragma unroll 4
  for (int kk = 0; kk < DM; kk += 4) {
    v2f a; a.x = xrow[kk];  a.y = xrow[kk + 1];            // A: 16x4 tile of x
    v2f bb;                                                // B: 4x16 tile of B^T
    bb.x = Bt[(kk + koff) * NS + mrow];
    bb.y = Bt[(kk + koff + 1) * NS + mrow];
    acc = wmma_f32_k4(a, bb, acc);
  }

  float cfb  = cf[b];
  int   half = lane >> 4;
  #pragma unroll
  for (int j = 0; j < 8; ++j) {
    int row = row0 + j + 8 * half;
    uB[(size_t)row * NS + mrow] = fast_tanh(acc[j] * cfb);
  }
}

// 5) serial recurrence: h_t = tanh(h_{t-1} @ A) + uB_t
//    One 16-lane group per batch. uB load software-pipelined one step ahead so
//    only {16 shfl-broadcasts -> 2 split FMA chains -> v_tanh} sit on the
//    4096-step dependent chain.
__global__ __launch_bounds__(16) void recur_kernel(const float* __restrict__ Aw,
                                                   const float* __restrict__ uB,
                                                   float* __restrict__ hs) {
  int b = blockIdx.x;
  int n = threadIdx.x;                             // 0..15
  float a[16];
  #pragma unroll
  for (int m = 0; m < 16; ++m) a[m] = Aw[m * 16 + n];      // column n of A

  const float* uBb = uB + (size_t)b * SEQ * NS;
  float*       hsb = hs + (size_t)b * SEQ * NS;

  float h = 0.f;
  float u = uBb[n];                                // u for t=0
  #pragma unroll 4
  for (int t = 0; t < SEQ; ++t) {
    // prefetch next step's injection off the critical path
    float u_next = (t + 1 < SEQ) ? uBb[(t + 1) * NS + n] : 0.f;
    if ((t & 31) == 0 && t + 32 < SEQ)
      __builtin_prefetch(&uBb[(t + 32) * NS], 0, 0);       // global_prefetch_b8

    float acc0 = 0.f, acc1 = 0.f;                          // split dep-chain
    #pragma unroll
    for (int m = 0; m < 16; ++m) {
      float hm = __shfl(h, m, 32);                         // broadcast h[m]
      if (m < 8) acc0 = fmaf(hm, a[m], acc0);
      else       acc1 = fmaf(hm, a[m], acc1);
    }
    h = fast_tanh(acc0 + acc1) + u;
    hsb[t * NS + n] = h;                                   // coalesced 64B
    u = u_next;
  }
}

// 6) y = tanh(hs @ C) + D * cf * x   via WMMA (K=16 -> 4 chained WMMAs)
//    Block owns one 16-row tile; each wave covers 8 of the 64 column tiles.
__global__ __launch_bounds__(256) void out_gemm_kernel(const float* __restrict__ hs,
                                                       const float* __restrict__ Cw,
                                                       const float* __restrict__ x,
                                                       const float* __restrict__ Dv,
                                                       const float* __restrict__ cf,
                                                       float* __restrict__ y) {
  int tid = threadIdx.x, wave = tid >> 5, lane = tid & 31;
  int tile = blockIdx.x;                                   // 0..1023
  int row0 = tile * 16;
  int b = row0 / SEQ;
  float cfb = cf[b];
  int mrow = lane & 15;
  int koff = (lane >> 4) * 2;
  int half = lane >> 4;

  // A-frags for the full K=16 (4 chunks), loaded once, reused for 8 col tiles
  const float* hrow = hs + (size_t)(row0 + mrow) * NS + koff;
  v2f a0, a1, a2, a3;
  a0.x = hrow[0];  a0.y = hrow[1];
  a1.x = hrow[4];  a1.y = hrow[5];
  a2.x = hrow[8];  a2.y = hrow[9];
  a3.x = hrow[12]; a3.y = hrow[13];

  for (int ct = wave; ct < DM / 16; ct += 8) {
    int d0 = ct * 16;
    v8f acc = {};
    v2f bb;
    bb.x = Cw[(0  + koff) * DM + d0 + mrow]; bb.y = Cw[(1  + koff) * DM + d0 + mrow];
    acc = wmma_f32_k4(a0, bb, acc);
    bb.x = Cw[(4  + koff) * DM + d0 + mrow]; bb.y = Cw[(5  + koff) * DM + d0 + mrow];
    acc = wmma_f32_k4(a1, bb, acc);
    bb.x = Cw[(8  + koff) * DM + d0 + mrow]; bb.y = Cw[(9  + koff) * DM + d0 + mrow];
    acc = wmma_f32_k4(a2, bb, acc);
    bb.x = Cw[(12 + koff) * DM + d0 + mrow]; bb.y = Cw[(13 + koff) * DM + d0 + mrow];
    acc = wmma_f32_k4(a3, bb, acc);

    float dv = Dv[d0 + mrow] * cfb;
    #pragma unroll
    for (int j = 0; j < 8; ++j) {
      int row = row0 + j + 8 * half;
      size_t idx = (size_t)row * DM + d0 + mrow;           // 64B-contiguous/16 lanes
      y[idx] = fast_tanh(acc[j]) + dv * x[idx];
    }
  }
}

// ---------------------------------------------------------------------------
extern "C" void kernel_launch(void* const* d_in, const int* in_sizes, int n_in,
                              void* d_out, int out_size, void* d_ws, size_t ws_size,
                              hipStream_t stream) {
  const float* x  = (const float*)d_in[0];
  const float* A  = (const float*)d_in[1];
  const float* B  = (const float*)d_in[2];
  const float* C  = (const float*)d_in[3];
  const float* D  = (const float*)d_in[4];
  const float* w1 = (const float*)d_in[5];
  const float* b1 = (const float*)d_in[6];
  const float* w2 = (const float*)d_in[7];
  const float* b2 = (const float*)d_in[8];
  float* y = (float*)d_out;

  // workspace layout (bytes)
  char* ws = (char*)d_ws;
  float* xpart = (float*)(ws);                               // 128*1024 f32 (512KB)
  float* xsum  = (float*)(ws + 524288);                      // 4*1024 f32
  float* cf    = (float*)(ws + 540672);                      // 4 f32
  float* uB    = (float*)(ws + 557056);                      // 16384*16 f32 (1MB)
  float* hsb   = (float*)(ws + 557056 + (size_t)NROW*NS*4);  // 1MB

  mean_kernel  <<<B_SZ * SPLIT, 256, 0, stream>>>(x, xpart);
  reduce_kernel<<<(B_SZ * DM + 255) / 256, 256, 0, stream>>>(xpart, xsum);
  gate_kernel  <<<1, 256, 0, stream>>>(xsum, w1, b1, w2, b2, cf);
  ub_gemm_kernel<<<NROW / 16 / 4, 128, 0, stream>>>(x, B, cf, uB);
  recur_kernel <<<B_SZ, 16, 0, stream>>>(A, uB, hsb);
  out_gemm_kernel<<<NROW / 16, 256, 0, stream>>>(hsb, C, x, D, cf, y);
}